// GNN_G_87093346828668
// MI455X (gfx1250) — compile-verified
//
#include <hip/hip_runtime.h>
#include <math.h>

typedef float v2f __attribute__((ext_vector_type(2)));
typedef float v8f __attribute__((ext_vector_type(8)));

#define NN    500   // NUM_NODES
#define NPAD  512
#define INC   32    // IN_C
#define HID   64
#define BSZ   32    // batch
#define TSZ   24    // time

// D = A(16x4) * B(4x16) + C, all fp32 (native CDNA5 fp32 matrix op)
static __device__ __forceinline__ v8f wmma4(v2f a, v2f b, v8f c) {
  return __builtin_amdgcn_wmma_f32_16x16x4_f32(
      /*neg_a=*/false, a, /*neg_b=*/false, b,
      /*c_mod=*/(short)0, c, /*reuse_a=*/false, /*reuse_b=*/false);
}

// ---------------------------------------------------------------------------
// Adjacency: A[n][m] = thresh(softmax_m(relu(e1[n]·e2[m]))), row-major, NPAD
// stride. Rows n in [500,512) zeroed so padded B-fragment reads contribute 0.
// ---------------------------------------------------------------------------
__global__ void k_adj(const float* __restrict__ e1, const float* __restrict__ e2,
                      float* __restrict__ A) {
  const int n   = blockIdx.x;
  const int tid = threadIdx.x;
  __shared__ float red[256];
  if (n >= NN) {
    for (int m = tid; m < NN; m += 256) A[(size_t)n * NPAD + m] = 0.0f;
    return;
  }
  const int m0 = tid, m1 = tid + 256;
  const bool h0 = (m0 < NN), h1 = (m1 < NN);
  float z0 = -1e30f, z1 = -1e30f;
  if (h0) {
    float d = 0.f;
#pragma unroll 8
    for (int k = 0; k < HID; ++k) d += e1[n * HID + k] * e2[m0 * HID + k];
    z0 = fmaxf(d, 0.f);
  }
  if (h1) {
    float d = 0.f;
#pragma unroll 8
    for (int k = 0; k < HID; ++k) d += e1[n * HID + k] * e2[m1 * HID + k];
    z1 = fmaxf(d, 0.f);
  }
  red[tid] = fmaxf(z0, z1);
  __syncthreads();
  for (int s = 128; s > 0; s >>= 1) {
    if (tid < s) red[tid] = fmaxf(red[tid], red[tid + s]);
    __syncthreads();
  }
  const float mx = red[0];
  __syncthreads();
  const float p0 = h0 ? expf(z0 - mx) : 0.f;
  const float p1 = h1 ? expf(z1 - mx) : 0.f;
  red[tid] = p0 + p1;
  __syncthreads();
  for (int s = 128; s > 0; s >>= 1) {
    if (tid < s) red[tid] += red[tid + s];
    __syncthreads();
  }
  const float inv = 1.0f / red[0];
  if (h0) { float a = p0 * inv; A[(size_t)n * NPAD + m0] = (a > 0.2f) ? a : 0.f; }
  if (h1) { float a = p1 * inv; A[(size_t)n * NPAD + m1] = (a > 0.2f) ? a : 0.f; }
}

// ---------------------------------------------------------------------------
// First conv on the t = T-1 slice only:
//   cur[b][o][n] = sum_c w_first[o][c] * x[b][c][n][T-1] + b_first[o]
// Block = 128 threads (4 waves); wave w owns output-channel tile w.
// ---------------------------------------------------------------------------
__global__ void k_first(const float* __restrict__ x, const float* __restrict__ wf,
                        const float* __restrict__ bf, float* __restrict__ cur,
                        float* __restrict__ oacc) {
  const int nt = blockIdx.x, b = blockIdx.y;
  const int wave = threadIdx.x >> 5;
  const int lane = threadIdx.x & 31;
  const int half = lane >> 4;
  const int row  = lane & 15;                 // M row for A-frag, N col for B-frag
  const int ncol = nt * 16 + row;
  const int ncl  = (ncol < NN) ? ncol : (NN - 1);   // clamp: keep EXEC all-ones
  const v2f* __restrict__ wfR = (const v2f*)(wf + (wave * 16 + row) * INC);
  v8f acc = {0.f, 0.f, 0.f, 0.f, 0.f, 0.f, 0.f, 0.f};
#pragma unroll
  for (int ks = 0; ks < INC / 4; ++ks) {
    const int kb = ks * 4 + half * 2;
    v2f a = wfR[ks * 2 + half];
    v2f bb;
    bb.x = x[(((size_t)b * INC + kb)     * NN + ncl) * TSZ + (TSZ - 1)];
    bb.y = x[(((size_t)b * INC + kb + 1) * NN + ncl) * TSZ + (TSZ - 1)];
    acc = wmma4(a, bb, acc);
  }
  if (ncol < NN) {
#pragma unroll
    for (int r = 0; r < 8; ++r) {
      const int o = wave * 16 + r + half * 8;
      const float v = acc[r] + bf[o];
      const size_t idx = ((size_t)b * HID + o) * NPAD + ncol;
      cur[idx]  = v;
      oacc[idx] = v;   // out = h
    }
  }
}

// ---------------------------------------------------------------------------
// One GCN layer, fused, TWO n-tiles per block (cur fragment reused 2x):
//   agg[c][n] = sum_m cur[b][c][m] * A[n][m]   (K = 500, 125 fp32-WMMA steps)
//   new = W_i @ agg + b_i + cur ;  oacc += new
// Block = 128 threads (4 waves); wave w owns channel tile w for both n-tiles.
// ---------------------------------------------------------------------------
__global__ void k_layer(const float* __restrict__ curi, const float* __restrict__ Aadj,
                        const float* __restrict__ wg, const float* __restrict__ bg,
                        float* __restrict__ curo, float* __restrict__ oacc) {
  __shared__ float aggS[HID * 32];   // [c][n_local 0..31]
  const int nt = blockIdx.x, b = blockIdx.y;       // nt covers 32 node columns
  const int wave = threadIdx.x >> 5;
  const int lane = threadIdx.x & 31;
  const int half = lane >> 4;
  const int row  = lane & 15;
  const int n0 = nt * 32 + row;
  const int n1 = n0 + 16;

  // ---- aggregation GEMM: one cur-frag load feeds two WMMAs ----
  const v2f* __restrict__ curR = (const v2f*)(curi + ((size_t)b * HID + wave * 16 + row) * NPAD);
  const v2f* __restrict__ aR0  = (const v2f*)(Aadj + (size_t)n0 * NPAD);  // rows>=500 zero
  const v2f* __restrict__ aR1  = (const v2f*)(Aadj + (size_t)n1 * NPAD);
  v8f acc0 = {0.f, 0.f, 0.f, 0.f, 0.f, 0.f, 0.f, 0.f};
  v8f acc1 = {0.f, 0.f, 0.f, 0.f, 0.f, 0.f, 0.f, 0.f};
  for (int ks = 0; ks < NN / 4; ++ks) {
    const int ki = ks * 2 + half;      // v2f index == (ks*4 + half*2)/2
    v2f a  = curR[ki];
    v2f b0 = aR0[ki];
    v2f b1 = aR1[ki];
    acc0 = wmma4(a, b0, acc0);
    acc1 = wmma4(a, b1, acc1);
  }
  // stage agg tiles (64 x 32) in LDS
#pragma unroll
  for (int r = 0; r < 8; ++r) {
    const int c = wave * 16 + r + half * 8;
    aggS[c * 32 + row]      = acc0[r];
    aggS[c * 32 + 16 + row] = acc1[r];
  }
  __syncthreads();

  // ---- channel-mix GEMM: new = W(64x64) @ agg(64x32), K = 64 ----
  const v2f* __restrict__ wgR = (const v2f*)(wg + (wave * 16 + row) * HID);
  v8f d0 = {0.f, 0.f, 0.f, 0.f, 0.f, 0.f, 0.f, 0.f};
  v8f d1 = {0.f, 0.f, 0.f, 0.f, 0.f, 0.f, 0.f, 0.f};
#pragma unroll
  for (int ks = 0; ks < HID / 4; ++ks) {
    const int kb = ks * 4 + half * 2;
    v2f a = wgR[ks * 2 + half];
    v2f b0, b1;
    b0.x = aggS[(kb)     * 32 + row];
    b0.y = aggS[(kb + 1) * 32 + row];
    b1.x = aggS[(kb)     * 32 + 16 + row];
    b1.y = aggS[(kb + 1) * 32 + 16 + row];
    d0 = wmma4(a, b0, d0);
    d1 = wmma4(a, b1, d1);
  }
  if (n0 < NN) {
#pragma unroll
    for (int r = 0; r < 8; ++r) {
      const int o = wave * 16 + r + half * 8;
      const size_t idx = ((size_t)b * HID + o) * NPAD + n0;
      const float v = d0[r] + bg[o] + curi[idx];   // + residual
      curo[idx]  = v;
      oacc[idx] += v;                              // out += new
    }
  }
  if (n1 < NN) {
#pragma unroll
    for (int r = 0; r < 8; ++r) {
      const int o = wave * 16 + r + half * 8;
      const size_t idx = ((size_t)b * HID + o) * NPAD + n1;
      const float v = d1[r] + bg[o] + curi[idx];
      curo[idx]  = v;
      oacc[idx] += v;
    }
  }
}

// ---------------------------------------------------------------------------
// Head: sigmoid( w_last · leaky_relu(out_acc[b,:,n]) + b_last )
// ---------------------------------------------------------------------------
__global__ void k_final(const float* __restrict__ oacc, const float* __restrict__ wl,
                        const float* __restrict__ bl, float* __restrict__ out) {
  const int idx = blockIdx.x * blockDim.x + threadIdx.x;
  if (idx >= BSZ * NN) return;
  const int b = idx / NN, n = idx % NN;
  float s = bl[0];
#pragma unroll 8
  for (int c = 0; c < HID; ++c) {
    float v = oacc[((size_t)b * HID + c) * NPAD + n];
    v = (v > 0.f) ? v : 0.01f * v;   // leaky_relu(0.01)
    s += wl[c] * v;
  }
  out[idx] = 1.0f / (1.0f + expf(-s));
}

// ---------------------------------------------------------------------------
extern "C" void kernel_launch(void* const* d_in, const int* in_sizes, int n_in,
                              void* d_out, int out_size, void* d_ws, size_t ws_size,
                              hipStream_t stream) {
  const float* x   = (const float*)d_in[0];
  const float* wf  = (const float*)d_in[1];
  const float* bf  = (const float*)d_in[2];
  const float* e1  = (const float*)d_in[3];
  const float* e2  = (const float*)d_in[4];
  const float* wg  = (const float*)d_in[5];   // (4,64,64)
  const float* bg  = (const float*)d_in[6];   // (4,64)
  const float* wl  = (const float*)d_in[7];
  const float* bl  = (const float*)d_in[8];
  float* out = (float*)d_out;

  float* ws   = (float*)d_ws;
  float* A    = ws;                                   // 512*512
  float* cur0 = A    + (size_t)NPAD * NPAD;           // 32*64*512
  float* cur1 = cur0 + (size_t)BSZ * HID * NPAD;
  float* oacc = cur1 + (size_t)BSZ * HID * NPAD;      // total ~13.6 MB

  k_adj<<<NPAD, 256, 0, stream>>>(e1, e2, A);
  k_first<<<dim3(32, BSZ), 128, 0, stream>>>(x, wf, bf, cur0, oacc);
  k_layer<<<dim3(16, BSZ), 128, 0, stream>>>(cur0, A, wg + 0 * HID * HID, bg + 0 * HID, cur1, oacc);
  k_layer<<<dim3(16, BSZ), 128, 0, stream>>>(cur1, A, wg + 1 * HID * HID, bg + 1 * HID, cur0, oacc);
  k_layer<<<dim3(16, BSZ), 128, 0, stream>>>(cur0, A, wg + 2 * HID * HID, bg + 2 * HID, cur1, oacc);
  k_layer<<<dim3(16, BSZ), 128, 0, stream>>>(cur1, A, wg + 3 * HID * HID, bg + 3 * HID, cur0, oacc);
  k_final<<<(BSZ * NN + 255) / 256, 256, 0, stream>>>(oacc, wl, bl, out);
}